// OGCNConv_10496900071975
// MI455X (gfx1250) — compile-verified
//
#include <hip/hip_runtime.h>
#include <math.h>

typedef float v2f __attribute__((ext_vector_type(2)));
typedef float v8f __attribute__((ext_vector_type(8)));

#define D 128

// ---------------------------------------------------------------------------
// WMMA f32 GEMM: C[n,128] = A[n,128] @ W[128,128]  (+ init modes)
// mode 0: C = A@W ; mode 1: C += A@W ; mode 2: C = broadcast(initRow) + A@W
// W is staged once per block into LDS, transposed + swizzled so each B
// fragment is a single conflict-free ds_load_b64. One wave per 16-row tile,
// 8 waves per block, V_WMMA_F32_16X16X4_F32 with A-fragment reuse over 8
// N-tiles (256 WMMA per wave).
// ---------------------------------------------------------------------------
__global__ __launch_bounds__(256)
void gemm128_wmma(const float* __restrict__ A,
                  const float* __restrict__ W,
                  float* __restrict__ C,
                  const float* __restrict__ initRow,
                  int n, int mode) {
  __shared__ float Wt[D * D];                 // 64 KB (WGP has 320 KB)

  // Stage W transposed: logical Wt[col][k] = W[k][col], with rotation
  // swizzle k -> (k + 4*(col&15)) & 127 (keeps even/odd k pairs contiguous,
  // spreads a wave's b64 reads across all 64 banks).
  for (int i = threadIdx.x; i < D * D; i += 256) {
    const int k = i >> 7, col = i & 127;
    Wt[col * D + ((k + 4 * (col & 15)) & 127)] = W[i];
  }
  __syncthreads();

  const int wave = threadIdx.x >> 5;
  const int lane = threadIdx.x & 31;
  const int ntiles = (n + 15) >> 4;
  const int tile = blockIdx.x * 8 + wave;
  if (tile >= ntiles) return;                 // wave-uniform: EXEC stays all-1s

  const int row0 = tile << 4;
  const int ml = lane & 15;                   // M (A) / N (B,C) within tile
  const int hi = lane >> 4;                   // half-wave select
  const int kbase = hi << 1;                  // K offset 0 or 2 (ISA A/B layout)
  const int rot = 4 * ml;                     // per-lane LDS swizzle rotation

  int m = row0 + ml;
  if (m > n - 1) m = n - 1;                   // clamp A reads; stores guarded
  const float* __restrict__ arow = A + (size_t)m * D;

  v8f acc[8];
  if (mode == 1) {
#pragma unroll
    for (int j = 0; j < 8; ++j) {
      const int col = j * 16 + ml;
#pragma unroll
      for (int v = 0; v < 8; ++v) {
        int rm = row0 + v + hi * 8;
        if (rm > n - 1) rm = n - 1;
        acc[j][v] = C[(size_t)rm * D + col];
      }
    }
  } else if (mode == 2) {
#pragma unroll
    for (int j = 0; j < 8; ++j) {
      const float rv = initRow[j * 16 + ml];
#pragma unroll
      for (int v = 0; v < 8; ++v) acc[j][v] = rv;
    }
  } else {
#pragma unroll
    for (int j = 0; j < 8; ++j)
#pragma unroll
      for (int v = 0; v < 8; ++v) acc[j][v] = 0.0f;
  }

  for (int k0 = 0; k0 < D; k0 += 4) {
    // A fragment: lane holds A[m][k0+kbase], A[m][k0+kbase+1] (one b64 load)
    const v2f a = *(const v2f*)(arow + k0 + kbase);
    const int krot = (k0 + kbase + rot) & 127; // even, +1 never wraps
#pragma unroll
    for (int j = 0; j < 8; ++j) {
      const int col = j * 16 + ml;
      // B fragment: W[k][col], W[k+1][col] -> one conflict-free ds_load_b64
      const v2f b = *(const v2f*)(&Wt[col * D + krot]);
      acc[j] = __builtin_amdgcn_wmma_f32_16x16x4_f32(
          false, a, false, b, (short)0, acc[j], false, false);
    }
  }

#pragma unroll
  for (int j = 0; j < 8; ++j) {
    const int col = j * 16 + ml;
#pragma unroll
    for (int v = 0; v < 8; ++v) {
      const int rm = row0 + v + hi * 8;
      if (rm < n) C[(size_t)rm * D + col] = acc[j][v];
    }
  }
}

// ---------------------------------------------------------------------------
// Degree / norm kernels
// ---------------------------------------------------------------------------
__global__ void deg_init(float* __restrict__ d0, float* __restrict__ d1, int n) {
  int i = blockIdx.x * blockDim.x + threadIdx.x;
  if (i < n) { d0[i] = 1.0f; d1[i] = 1.0f; }   // self-loop weight 1
}

__global__ void deg_accum(const int* __restrict__ ei, const int* __restrict__ et,
                          float* __restrict__ d0, float* __restrict__ d1, int E) {
  int e = blockIdx.x * blockDim.x + threadIdx.x;
  if (e >= E) return;
  int c = ei[E + e];                            // col
  float* d = et[e] ? d1 : d0;
  atomicAdd(&d[c], 1.0f);
}

__global__ void make_dinv(float* __restrict__ d0, float* __restrict__ d1, int n) {
  int i = blockIdx.x * blockDim.x + threadIdx.x;
  if (i < n) { d0[i] = rsqrtf(d0[i]); d1[i] = rsqrtf(d1[i]); }  // deg >= 1 always
}

// Self-loop contribution: X_t[i] = H_t[i] * dinv_t[i]^2
__global__ void self_init(const float* __restrict__ H0, const float* __restrict__ H1,
                          const float* __restrict__ v0, const float* __restrict__ v1,
                          float* __restrict__ Xn, float* __restrict__ Xd, int n) {
  size_t idx = (size_t)blockIdx.x * blockDim.x + threadIdx.x;
  if (idx >= (size_t)n * D) return;
  int i = (int)(idx >> 7);
  float s0 = v0[i], s1 = v1[i];
  Xn[idx] = H0[idx] * s0 * s0;
  Xd[idx] = H1[idx] * s1 * s1;
}

// Edge scatter: 128 threads per edge; coalesced gather + f32 global atomics
// (sources + destinations ~102 MB total -> resident in 192 MB L2)
__global__ __launch_bounds__(256)
void scatter_edges(const int* __restrict__ ei, const int* __restrict__ et,
                   const float* __restrict__ H0, const float* __restrict__ H1,
                   const float* __restrict__ v0, const float* __restrict__ v1,
                   float* __restrict__ Xn, float* __restrict__ Xd, int E) {
  int e = blockIdx.x * 2 + (threadIdx.x >> 7);
  if (e >= E) return;
  int d = threadIdx.x & 127;
  int r = ei[e];
  int c = ei[E + e];
  int t = et[e];
  const float* __restrict__ H = t ? H1 : H0;
  const float* __restrict__ dv = t ? v1 : v0;
  float* __restrict__ X = t ? Xd : Xn;
  float norm = dv[r] * dv[c];
  atomicAdd(&X[(size_t)c * D + d], H[(size_t)r * D + d] * norm);
}

// Constant gate row: r = b_sl^T Wg0 + b0^T Wg1 + b1^T Wg2 + b_g
__global__ void gate_const_row(const float* __restrict__ bsl, const float* __restrict__ b0,
                               const float* __restrict__ b1, const float* __restrict__ Wg,
                               const float* __restrict__ bg, float* __restrict__ r) {
  int d = threadIdx.x;
  float s = bg[d];
  for (int j = 0; j < D; ++j) {
    s += bsl[j] * Wg[j * D + d];
    s += b0[j]  * Wg[(D + j) * D + d];
    s += b1[j]  * Wg[(2 * D + j) * D + d];
  }
  r[d] = s;
}

// Softmax -> cumsum -> combine. One 128-thread block per node.
// G lives in d_out; each thread reads only its own slot before overwriting.
__global__ __launch_bounds__(128)
void finalize(const float* __restrict__ Hsl, const float* __restrict__ Xn,
              const float* __restrict__ Xd,
              const float* __restrict__ bsl, const float* __restrict__ b0,
              const float* __restrict__ b1,
              const float* __restrict__ G, float* __restrict__ out, int n) {
  __shared__ float red[D];
  __shared__ float sm[D];
  int i = blockIdx.x;
  int d = threadIdx.x;
  float g = G[(size_t)i * D + d];

  red[d] = g; __syncthreads();
  for (int s = 64; s > 0; s >>= 1) {
    if (d < s) red[d] = fmaxf(red[d], red[d + s]);
    __syncthreads();
  }
  float mx = red[0]; __syncthreads();
  float e = expf(g - mx);
  red[d] = e; __syncthreads();
  for (int s = 64; s > 0; s >>= 1) {
    if (d < s) red[d] += red[d + s];
    __syncthreads();
  }
  float p = e / red[0]; __syncthreads();

  sm[d] = p; __syncthreads();                 // inclusive Hillis-Steele scan
  for (int off = 1; off < D; off <<= 1) {
    float v = (d >= off) ? sm[d - off] : 0.0f;
    __syncthreads();
    sm[d] += v;
    __syncthreads();
  }
  float gat = sm[d];

  float xx = Hsl[(size_t)i * D + d] + bsl[d];
  float xn = Xn[(size_t)i * D + d] + b0[d];
  float xdrev = Xd[(size_t)i * D + (D - 1 - d)] + b1[D - 1 - d];
  out[(size_t)i * D + d] = xdrev * gat + xx + xn;
}

// ---------------------------------------------------------------------------
extern "C" void kernel_launch(void* const* d_in, const int* in_sizes, int n_in,
                              void* d_out, int out_size, void* d_ws, size_t ws_size,
                              hipStream_t stream) {
  const float* x   = (const float*)d_in[0];
  const int* ei    = (const int*)d_in[1];     // [2*E] rows then cols
  const int* et    = (const int*)d_in[2];     // [E]
  const float* Wsl = (const float*)d_in[3];
  const float* bsl = (const float*)d_in[4];
  const float* W0  = (const float*)d_in[5];
  const float* b0  = (const float*)d_in[6];
  const float* W1  = (const float*)d_in[7];
  const float* b1  = (const float*)d_in[8];
  const float* Wg  = (const float*)d_in[9];
  const float* bg  = (const float*)d_in[10];

  const int N = in_sizes[0] / D;
  const int E = in_sizes[2];
  float* out = (float*)d_out;

  // workspace carve-up (floats)
  float* ws   = (float*)d_ws;
  size_t nd   = (size_t)N * D;
  float* Hsl  = ws;
  float* H0   = ws + nd;
  float* H1   = ws + 2 * nd;
  float* Xn   = ws + 3 * nd;
  float* Xd   = ws + 4 * nd;
  float* dg0  = ws + 5 * nd;
  float* dg1  = dg0 + N;
  float* crow = dg1 + N;                      // 128 floats

  const int ntiles = (N + 15) / 16;
  const int gemmBlocks = (ntiles + 7) / 8;

  // 1) degrees + symmetric norms
  deg_init<<<(N + 255) / 256, 256, 0, stream>>>(dg0, dg1, N);
  deg_accum<<<(E + 255) / 256, 256, 0, stream>>>(ei, et, dg0, dg1, E);
  make_dinv<<<(N + 255) / 256, 256, 0, stream>>>(dg0, dg1, N);

  // 2) feature GEMMs (fp32 WMMA, LDS-staged weights)
  gemm128_wmma<<<gemmBlocks, 256, 0, stream>>>(x, Wsl, Hsl, nullptr, N, 0);
  gemm128_wmma<<<gemmBlocks, 256, 0, stream>>>(x, W0,  H0,  nullptr, N, 0);
  gemm128_wmma<<<gemmBlocks, 256, 0, stream>>>(x, W1,  H1,  nullptr, N, 0);

  // 3) aggregation: self-loops then typed-edge scatter
  self_init<<<(unsigned)((nd + 255) / 256), 256, 0, stream>>>(H0, H1, dg0, dg1, Xn, Xd, N);
  scatter_edges<<<(E + 1) / 2, 256, 0, stream>>>(ei, et, H0, H1, dg0, dg1, Xn, Xd, E);

  // 4) gate GEMM: G = [Hsl|Xn|Xd] @ Wg + (bias-induced constant row), G -> d_out
  gate_const_row<<<1, D, 0, stream>>>(bsl, b0, b1, Wg, bg, crow);
  gemm128_wmma<<<gemmBlocks, 256, 0, stream>>>(Hsl, Wg,             out, crow, N, 2);
  gemm128_wmma<<<gemmBlocks, 256, 0, stream>>>(Xn,  Wg + D * D,     out, nullptr, N, 1);
  gemm128_wmma<<<gemmBlocks, 256, 0, stream>>>(Xd,  Wg + 2 * D * D, out, nullptr, N, 1);

  // 5) softmax + cumsum gate + combine
  finalize<<<N, D, 0, stream>>>(Hsl, Xn, Xd, bsl, b0, b1, out, out, N);
}